// subbands_to_envelopes_4475355922983
// MI455X (gfx1250) — compile-verified
//
#include <hip/hip_runtime.h>
#include <stdint.h>

typedef __attribute__((ext_vector_type(2))) float v2f;
typedef __attribute__((ext_vector_type(8))) float v8f;

// Problem constants (fixed by the reference).
#define T_IN    40000
#define T_OUT   20000
#define NTAPS   41
#define PAD     20                       // (NTAPS-1)/2
#define TILE    2048                     // outputs per block (8 waves x 256)
#define TPB     256
#define IN_ELEMS (2 * TILE + NTAPS - 2)  // 4135 input samples per tile
#define NCHUNK   ((IN_ELEMS + 3) / 4)    // 1034 x 16B async chunks
#define SMEM_ELEMS (NCHUNK * 4)          // 4136 floats (max index used = 4135)
#define KSTEPS  18                       // K = 72 (banded width 71 padded to 72)
#define BROWS   (2 * KSTEPS)             // 36 packed float2 rows of B

__global__ __launch_bounds__(TPB)
void envconv_wmma_kernel(const float* __restrict__ x,
                         const float* __restrict__ filt,
                         float* __restrict__ out) {
    __shared__ __align__(16) float sin_[SMEM_ELEMS];
    __shared__ __align__(8)  v2f   sB[BROWS][16];   // B[j,r]=h[j-2r], packed (j, j+1)

    const int tid  = threadIdx.x;
    const int tile = blockIdx.x;                 // 0..9
    const int row  = blockIdx.y;                 // 0..799 (B*F folded)
    const int o0   = tile * TILE;
    const int s0   = 2 * o0 - PAD;               // first global input index (multiple of 4)

    const uint64_t rowbase = (uint64_t)(uintptr_t)(x + (size_t)row * T_IN);

    // ---- Stage input tile into LDS with CDNA5 async copies (16B per chunk) ----
    for (int c = tid; c < NCHUNK; c += TPB) {
        int gc  = s0 + 4 * c;                                       // global chunk start
        int gcl = gc < 0 ? 0 : (gc > (T_IN - 4) ? (T_IN - 4) : gc); // clamp, stays 16B aligned
        uint32_t ldsb = (uint32_t)(uintptr_t)(&sin_[4 * c]);        // LDS byte offset (addr[31:0])
        uint32_t goff = (uint32_t)(gcl * 4);
        asm volatile("global_load_async_to_lds_b128 %0, %1, %2"
                     :
                     : "v"(ldsb), "v"(goff), "s"(rowbase)
                     : "memory");
    }

    // ---- Build banded B matrix from the 41 taps: B[j,r] = h[j-2r] (else 0) ----
    for (int idx = tid; idx < BROWS * 16; idx += TPB) {
        int t2 = idx >> 4;            // packed row pair index: j = 2*t2, 2*t2+1
        int n  = idx & 15;            // r
        int k0 = 2 * t2     - 2 * n;
        int k1 = 2 * t2 + 1 - 2 * n;
        v2f b;
        b.x = (k0 >= 0 && k0 < NTAPS) ? filt[k0] : 0.0f;
        b.y = (k1 >= 0 && k1 < NTAPS) ? filt[k1] : 0.0f;
        sB[t2][n] = b;
    }

    asm volatile("s_wait_asynccnt 0" ::: "memory");
    __syncthreads();

    // ---- One-pass input relu + zero out-of-range (padded/clamped) samples ----
    for (int j = tid; j < SMEM_ELEMS; j += TPB) {
        int g = s0 + j;
        float v = sin_[j];
        sin_[j] = (g >= 0 && g < T_IN) ? fmaxf(v, 0.0f) : 0.0f;
    }
    __syncthreads();

    // ---- 16x16 output tile per wave via 18 chained V_WMMA_F32_16X16X4_F32 ----
    const int w    = tid >> 5;             // wave id 0..7
    const int lane = tid & 31;
    const int m    = lane & 15;            // M (A-row / D-col index role per layout)
    const int hb   = lane >> 4;            // half-wave: K pair select
    const int kk   = hb << 1;              // 0 or 2
    const int abase = 512 * w + 32 * m + kk;

    v8f acc = {};
#pragma unroll
    for (int t = 0; t < KSTEPS; ++t) {
        // A: lane holds e[abase + 4t], e[abase + 4t + 1]  (aligned 8B LDS read)
        v2f a = *(const v2f*)&sin_[abase + 4 * t];
        // B: lane holds B[4t+kk, m], B[4t+kk+1, m] -> packed row 2t+hb, col m
        v2f b = sB[2 * t + hb][m];
        acc = __builtin_amdgcn_wmma_f32_16x16x4_f32(
                  /*neg_a=*/false, a, /*neg_b=*/false, b,
                  /*c_mod=*/(short)0, acc, /*reuse_a=*/false, /*reuse_b=*/false);
    }

    // ---- Store D: VGPR i, lane l -> output u = 256w + 16i + 128*(l>=16) + (l&15) ----
    float* orow = out + (size_t)row * T_OUT + o0;
    const int rbase = 256 * w + hb * 128 + m;
#pragma unroll
    for (int i = 0; i < 8; ++i) {
        int u = rbase + 16 * i;
        if (o0 + u < T_OUT) orow[u] = fmaxf(acc[i], 0.0f);
    }
}

extern "C" void kernel_launch(void* const* d_in, const int* in_sizes, int n_in,
                              void* d_out, int out_size, void* d_ws, size_t ws_size,
                              hipStream_t stream) {
    const float* x    = (const float*)d_in[0];   // [16,50,40000] fp32
    const float* filt = (const float*)d_in[1];   // [41] fp32
    float* out        = (float*)d_out;           // [16,50,20000] fp32

    const int rows  = in_sizes[0] / T_IN;              // 800
    const int tiles = (T_OUT + TILE - 1) / TILE;       // 10

    dim3 grid(tiles, rows);
    envconv_wmma_kernel<<<grid, TPB, 0, stream>>>(x, filt, out);
}